// GradientHistogramExtractor_50629074485728
// MI455X (gfx1250) — compile-verified
//
#include <hip/hip_runtime.h>

// GradientHistogramExtractor for MI455X (gfx1250, wave32).
// Pass 1: zero d_out (16384*1024 f32 histogram, used directly as accumulator).
// Pass 2: per-pixel bin compute + global_atomic_add_f32 scatter (L2-resident).
// Pass 3: one wave32 per segment; WMMA_F32_16X16X4_F32 ones-matmul reduction of
//         the 1024 bins -> segment size; in-place normalize.

typedef __attribute__((ext_vector_type(2))) float v2f;
typedef __attribute__((ext_vector_type(8))) float v8f;

#define PATCH   32
#define PP      1024        // PATCH*PATCH
#define HW_     262144      // 512*512 (fixed by reference)
#define EPSF    1e-7f

// ---------------- Pass 1: zero the histogram ----------------
__global__ void ghe_zero_kernel(float4* __restrict__ out, int n4) {
    int i = blockIdx.x * blockDim.x + threadIdx.x;
    if (i < n4) out[i] = make_float4(0.f, 0.f, 0.f, 0.f);
}

// ---------------- Pass 2: scatter-add histogram ----------------
__global__ void ghe_hist_kernel(const float* __restrict__ grad,
                                const int*   __restrict__ seg,
                                float*       __restrict__ hist,
                                int nPix4) {
    int q = blockIdx.x * blockDim.x + threadIdx.x;
    if (q >= nPix4) return;
    int p  = q << 2;               // first pixel of this group of 4
    int b  = p / HW_;              // batch index (HW_ divisible by 4 -> same b)
    int hw = p - b * HW_;

    const float4 gy4 = *(const float4*)(grad + (size_t)(2 * b)     * HW_ + hw);
    const float4 gx4 = *(const float4*)(grad + (size_t)(2 * b + 1) * HW_ + hw);
    const int4   s4  = *(const int4*)(seg + p);

    const float lo = EPSF - 1.0f, hi = 1.0f - EPSF;
    const float* gy = (const float*)&gy4;
    const float* gx = (const float*)&gx4;
    const int*   ss = (const int*)&s4;
#pragma unroll
    for (int j = 0; j < 4; ++j) {
        // g = PATCH * (clip(g) + 1) * 0.5, floor -> bin
        float cy = fminf(fmaxf(gy[j], lo), hi);
        float cx = fminf(fmaxf(gx[j], lo), hi);
        int iy = (int)floorf((float)PATCH * (cy + 1.0f) * 0.5f);
        int ix = (int)floorf((float)PATCH * (cx + 1.0f) * 0.5f);
        iy = min(max(iy, 0), PATCH - 1);
        ix = min(max(ix, 0), PATCH - 1);
        int pos = ss[j] * PP + iy * PATCH + ix;
        atomicAdd(hist + pos, 1.0f);   // -> global_atomic_add_f32 (no return)
    }
}

// ---------------- Pass 3: WMMA segment-sum + normalize ----------------
// One wave32 per segment. Each lane owns 32 of the segment's 1024 bins.
// Reduction: D = A(ones 16x4) x B(counts 4x16) + C, chained over 16 chunks.
// D[i][j] = column-sum_j for every i; c[0] lanes 0..15 hold the 16 distinct
// column sums (lanes 16..31 replicate them), so full-wave sum = 2 * total.
__global__ void ghe_norm_kernel(float* __restrict__ hist, int nSeg) {
    int wave = (blockIdx.x * blockDim.x + threadIdx.x) >> 5;
    if (wave >= nSeg) return;                 // grid sized exactly; keeps EXEC full
    int lane = threadIdx.x & 31;
    float* base = hist + (size_t)wave * PP;

    float vals[32];
#pragma unroll
    for (int ch = 0; ch < 8; ++ch) {
        float4 v = *(const float4*)(base + ch * 128 + lane * 4);
        vals[ch * 4 + 0] = v.x; vals[ch * 4 + 1] = v.y;
        vals[ch * 4 + 2] = v.z; vals[ch * 4 + 3] = v.w;
    }

    v2f a; a[0] = 1.0f; a[1] = 1.0f;          // ones A-matrix (16x4)
    v8f c = {};
#pragma unroll
    for (int i = 0; i < 16; ++i) {
        v2f bm; bm[0] = vals[2 * i]; bm[1] = vals[2 * i + 1];
        c = __builtin_amdgcn_wmma_f32_16x16x4_f32(
                /*neg_a=*/false, a, /*neg_b=*/false, bm,
                /*c_mod=*/(short)0, c, /*reuse_a=*/false, /*reuse_b=*/false);
    }

    float s = c[0];
#pragma unroll
    for (int m = 16; m >= 1; m >>= 1) s += __shfl_xor(s, m, 32);
    float total = s * 0.5f;                   // sizes[v]; den = sizes * (P/32)^2 = sizes
    float inv = 1.0f / total;

#pragma unroll
    for (int ch = 0; ch < 8; ++ch) {
        float4 v;
        v.x = vals[ch * 4 + 0] * inv; v.y = vals[ch * 4 + 1] * inv;
        v.z = vals[ch * 4 + 2] * inv; v.w = vals[ch * 4 + 3] * inv;
        *(float4*)(base + ch * 128 + lane * 4) = v;
    }
}

extern "C" void kernel_launch(void* const* d_in, const int* in_sizes, int n_in,
                              void* d_out, int out_size, void* d_ws, size_t ws_size,
                              hipStream_t stream) {
    const float* grad = (const float*)d_in[0];   // [B, 2, 512, 512] f32
    const int*   seg  = (const int*)d_in[1];     // [B, 512, 512] i32
    float*       out  = (float*)d_out;           // [nV, 1, 32, 32] f32

    const int nPix  = in_sizes[1];               // B*H*W
    const int nSeg  = out_size / PP;             // nV
    const int n4    = out_size / 4;
    const int nPix4 = nPix / 4;

    ghe_zero_kernel<<<(n4 + 255) / 256, 256, 0, stream>>>((float4*)out, n4);
    ghe_hist_kernel<<<(nPix4 + 255) / 256, 256, 0, stream>>>(grad, seg, out, nPix4);
    // 8 waves (8 segments) per 256-thread block
    ghe_norm_kernel<<<(nSeg + 7) / 8, 256, 0, stream>>>(out, nSeg);
    (void)d_ws; (void)ws_size; (void)n_in;
}